// APCNNCrop_50740743635440
// MI455X (gfx1250) — compile-verified
//
#include <hip/hip_runtime.h>
#include <hip/hip_bf16.h>
#include <math.h>

// ---------------------------------------------------------------------------
// APCNN attention-crop for MI455X (gfx1250, wave32).
// Memory-bound (~340 MB traffic -> ~15us @ 23.3 TB/s).
//  * channel mean over C=256 via chained V_WMMA_F32_16X16X4_F32 with B=ones:
//    D[m][n] = sum_k A[m][k]  (exact f32; 4 channels x 16 positions / WMMA).
//    HW is a template constant so all 128 channel loads per wave fold their
//    strides into the GLOBAL load's 24-bit immediate offset (max 3.2MB < 8MB)
//    -> long load clauses, deep LOADcnt pipelining, no per-load address VALU.
//  * NMS: one block per (level, batch), LDS scores, wave32 shfl_xor argmax
//    with min-index tie-break (= jnp.argmax first-occurrence).
//  * crop: bilinear gather; per-(n,c) source slice is 12.5KB -> cache-hot.
// ---------------------------------------------------------------------------

typedef __attribute__((ext_vector_type(2))) float v2f;
typedef __attribute__((ext_vector_type(8))) float v8f;

#define NEGV (-1e30f)
#define IMG_HW_CLAMP 447.0f

// ----------------------- channel mean via WMMA f32 -------------------------
// grid: (ceil(HW/16), N), block: 32 (one wave). Each wave owns 16 spatial
// positions and reduces all C channels with 64 chained WMMAs.
// A layout (16x4 f32): lanes 0-15 hold K=0,1 (VGPR0,1); lanes 16-31 hold
// K=2,3. M = lane & 15. B = ones -> row sums, replicated over N in D.
template <int HW, int C>
__global__ __launch_bounds__(32)
void channel_mean_wmma(const float* __restrict__ x, float* __restrict__ m) {
  const int tile = blockIdx.x;
  const int n    = blockIdx.y;
  const int lane = threadIdx.x;
  const int half = lane >> 4;     // 0: this lane supplies K=0,1   1: K=2,3
  const int row  = lane & 15;     // M index
  const int pos  = tile * 16 + row;
  const int posc = pos < HW ? pos : HW - 1;   // clamp (EXEC must stay all-1s)

  // single per-lane base; every channel access is a compile-time immediate
  const float* lanebase =
      x + (size_t)n * (size_t)C * (size_t)HW + (size_t)(2 * half) * HW + posc;

  v8f acc = {};
  v2f bones; bones.x = 1.0f; bones.y = 1.0f;
#pragma unroll
  for (int c = 0; c < C; c += 4) {
    v2f a;
    a.x = lanebase[(size_t)(c + 0) * HW];   // channel c + 2*half
    a.y = lanebase[(size_t)(c + 1) * HW];   // channel c + 2*half + 1
    // D = A x ones(4x16) + C  => per-row (per-position) partial channel sum
    acc = __builtin_amdgcn_wmma_f32_16x16x4_f32(false, a, false, bones,
                                                (short)0, acc, false, false);
  }
  // D layout: VGPR r = M=r (lanes 0-15) / M=8+r (lanes 16-31), same for all N.
  const float inv = 1.0f / (float)C;
  if (lane == 0 || lane == 16) {
    const int mbase = tile * 16 + (lane == 16 ? 8 : 0);
    float* mout = m + (size_t)n * HW;
#pragma unroll
    for (int r = 0; r < 8; ++r) {
      const int p = mbase + r;
      if (p < HW) mout[p] = acc[r] * inv;
    }
  }
}

// ------------------------------- NMS top-k ---------------------------------
// One block (128 threads = 4 waves) per batch element. Scores live in LDS.
// Argmax replicates jnp.argmax first-occurrence via min-index tie-break.
template <int H, int W, int TOPK>
__global__ __launch_bounds__(128)
void nms_kernel(const float* __restrict__ m, float* __restrict__ roi,
                float stridef, float sizef, float iou_thr) {
  const int n    = blockIdx.x;
  const int tid  = threadIdx.x;
  const int lane = tid & 31;
  const int wid  = tid >> 5;
  constexpr int HW = H * W;

  __shared__ float s[HW];
  __shared__ float rv[4];
  __shared__ int   ri[4];

  constexpr int h0 = (int)(0.1 * H), h1 = (int)(0.9 * H);
  constexpr int w0 = (int)(0.1 * W), w1 = (int)(0.9 * W);

  // corner-masked scores + partial sum
  float psum = 0.0f;
  for (int i = tid; i < HW; i += 128) {
    const int h = i / W, w = i - h * W;      // W is constexpr -> mul/shift
    float v = m[(size_t)n * HW + i];
    const bool in = (h >= h0) && (h < h1) && (w >= w0) && (w < w1);
    v = in ? v : 0.0f;
    s[i] = v;
    psum += v;
  }
  for (int off = 16; off > 0; off >>= 1) psum += __shfl_xor(psum, off, 32);
  if (lane == 0) rv[wid] = psum;
  __syncthreads();
  if (tid == 0) rv[0] = rv[0] + rv[1] + rv[2] + rv[3];
  __syncthreads();
  const float mean = rv[0] / (float)HW;
  __syncthreads();

  for (int i = tid; i < HW; i += 128) {
    const float v = s[i];
    s[i] = (v > mean) ? v : NEGV;
  }
  __syncthreads();

  const float halfs = 0.5f * sizef;
  const float area  = sizef * sizef;   // every anchor has identical area

  for (int k = 0; k < TOPK; ++k) {
    // local scan: ascending index + strict '>' keeps first occurrence
    float bv = -INFINITY;
    int   bi = 0x7fffffff;
    for (int i = tid; i < HW; i += 128) {
      const float v = s[i];
      if (v > bv) { bv = v; bi = i; }
    }
    for (int off = 16; off > 0; off >>= 1) {
      const float ov = __shfl_xor(bv, off, 32);
      const int   oi = __shfl_xor(bi, off, 32);
      if (ov > bv || (ov == bv && oi < bi)) { bv = ov; bi = oi; }
    }
    if (lane == 0) { rv[wid] = bv; ri[wid] = bi; }
    __syncthreads();
    if (tid == 0) {
      float cv = rv[0]; int ci = ri[0];
#pragma unroll
      for (int wv = 1; wv < 4; ++wv) {
        if (rv[wv] > cv || (rv[wv] == cv && ri[wv] < ci)) { cv = rv[wv]; ci = ri[wv]; }
      }
      rv[0] = cv; ri[0] = ci;
    }
    __syncthreads();
    const int idx = ri[0];
    __syncthreads();

    const int bh = idx / W, bw = idx - bh * W;
    const float bx1 = bw * stridef - halfs, by1 = bh * stridef - halfs;
    const float bx2 = bw * stridef + halfs, by2 = bh * stridef + halfs;

    if (tid == 0) {
      float* r = roi + (size_t)(n * TOPK + k) * 5;
      r[0] = (float)n;
      r[1] = fmaxf(bx1, 0.0f);
      r[2] = fmaxf(by1, 0.0f);
      r[3] = fminf(bx2, IMG_HW_CLAMP);
      r[4] = fminf(by2, IMG_HW_CLAMP);
    }

    // suppress IoU > thr against the UNclamped selected box
    for (int i = tid; i < HW; i += 128) {
      const int h = i / W, w = i - h * W;
      const float ax1 = w * stridef - halfs, ay1 = h * stridef - halfs;
      const float ax2 = w * stridef + halfs, ay2 = h * stridef + halfs;
      const float ix1 = fmaxf(bx1, ax1), iy1 = fmaxf(by1, ay1);
      const float ix2 = fminf(bx2, ax2), iy2 = fminf(by2, ay2);
      const float inter = fmaxf(ix2 - ix1, 0.0f) * fmaxf(iy2 - iy1, 0.0f);
      const float iou = inter / (area + area - inter);
      if (iou > iou_thr || i == idx) s[i] = NEGV;
    }
    __syncthreads();
  }
}

// ------------------------------- union box --------------------------------
__global__ void union_kernel(const float* __restrict__ roi3,
                             const float* __restrict__ roi4,
                             const float* __restrict__ roi5,
                             int* __restrict__ ub, int N) {
  const int n = blockIdx.x * blockDim.x + threadIdx.x;
  if (n >= N) return;
  float x1 = INFINITY, y1 = INFINITY, x2 = -INFINITY, y2 = -INFINITY;
#pragma unroll
  for (int k = 0; k < 5; ++k) {
    const float* r = roi3 + (size_t)(n * 5 + k) * 5;
    x1 = fminf(x1, r[1]); y1 = fminf(y1, r[2]);
    x2 = fmaxf(x2, r[3]); y2 = fmaxf(y2, r[4]);
  }
#pragma unroll
  for (int k = 0; k < 3; ++k) {
    const float* r = roi4 + (size_t)(n * 3 + k) * 5;
    x1 = fminf(x1, r[1]); y1 = fminf(y1, r[2]);
    x2 = fmaxf(x2, r[3]); y2 = fmaxf(y2, r[4]);
  }
  {
    const float* r = roi5 + (size_t)n * 5;
    x1 = fminf(x1, r[1]); y1 = fminf(y1, r[2]);
    x2 = fmaxf(x2, r[3]); y2 = fmaxf(y2, r[4]);
  }
  const float inv = 1.0f / 8.0f;
  const int xi1 = (int)floorf(x1 * inv);
  const int yi1 = (int)floorf(y1 * inv);
  const int xi2 = (int)floorf(x2 * inv);
  const int yi2 = (int)floorf(y2 * inv);
  ub[n * 4 + 0] = xi1;
  ub[n * 4 + 1] = yi1;
  ub[n * 4 + 2] = xi2 - xi1;   // cw
  ub[n * 4 + 3] = yi2 - yi1;   // ch
}

// --------------------------- bilinear crop-resize --------------------------
// grid: (C, N), block 256. Per-(n,c) source slice is 12.5KB -> the 4 bilinear
// taps hit cache; stores are fully coalesced.
template <int H, int W>
__global__ __launch_bounds__(256)
void crop_kernel(const float* __restrict__ x2, const int* __restrict__ ub,
                 float* __restrict__ out, int C) {
  const int c = blockIdx.x;
  const int n = blockIdx.y;
  const int x1i = ub[n * 4 + 0], y1i = ub[n * 4 + 1];
  const int cw  = ub[n * 4 + 2], ch  = ub[n * 4 + 3];
  const float sx = (float)cw / (float)W;
  const float sy = (float)ch / (float)H;
  const float* src = x2 + ((size_t)n * C + c) * (size_t)(H * W);
  float* dst       = out + ((size_t)n * C + c) * (size_t)(H * W);

  for (int p = threadIdx.x; p < H * W; p += 256) {
    const int oy = p / W, ox = p - oy * W;   // constexpr W -> mul/shift
    const float syf = fmaxf((oy + 0.5f) * sy - 0.5f, 0.0f);
    int i0y = (int)syf;                       // floor (syf >= 0)
    int i1y = i0y + 1; if (i1y > ch - 1) i1y = ch - 1; if (i1y < 0) i1y = 0;
    const float fy = syf - (float)i0y;
    const int gy0 = y1i + i0y, gy1 = y1i + i1y;

    const float sxf = fmaxf((ox + 0.5f) * sx - 0.5f, 0.0f);
    int i0x = (int)sxf;
    int i1x = i0x + 1; if (i1x > cw - 1) i1x = cw - 1; if (i1x < 0) i1x = 0;
    const float fx = sxf - (float)i0x;
    const int gx0 = x1i + i0x, gx1 = x1i + i1x;

    const float a = src[gy0 * W + gx0];
    const float b = src[gy0 * W + gx1];
    const float d = src[gy1 * W + gx0];
    const float e = src[gy1 * W + gx1];
    const float top = a + fx * (b - a);
    const float bot = d + fx * (e - d);
    dst[p] = top + fy * (bot - top);
  }
}

// ------------------------------- launcher ----------------------------------
extern "C" void kernel_launch(void* const* d_in, const int* in_sizes, int n_in,
                              void* d_out, int out_size, void* d_ws, size_t ws_size,
                              hipStream_t stream) {
  const float* x2 = (const float*)d_in[0];   // (32,256,56,56)
  const float* x3 = (const float*)d_in[1];   // (32,256,56,56)
  const float* x4 = (const float*)d_in[2];   // (32,256,28,28)
  const float* x5 = (const float*)d_in[3];   // (32,256,14,14)

  float* out  = (float*)d_out;
  float* roi3 = out;                              // (160,5)
  float* roi4 = out + 160 * 5;                    // (96,5)
  float* roi5 = out + 160 * 5 + 96 * 5;           // (32,5)
  float* crop = out + 160 * 5 + 96 * 5 + 32 * 5;  // (32,256,56,56)

  const int N = 32, C = 256;
  constexpr int HW3 = 56 * 56, HW4 = 28 * 28, HW5 = 14 * 14;

  // workspace: m3 | m4 | m5 | union boxes (ints)   (~527 KB)
  float* m3 = (float*)d_ws;
  float* m4 = m3 + (size_t)N * HW3;
  float* m5 = m4 + (size_t)N * HW4;
  int*   ub = (int*)(m5 + (size_t)N * HW5);

  dim3 wave(32, 1, 1);
  channel_mean_wmma<HW3, 256><<<dim3((HW3 + 15) / 16, N), wave, 0, stream>>>(x3, m3);
  channel_mean_wmma<HW4, 256><<<dim3((HW4 + 15) / 16, N), wave, 0, stream>>>(x4, m4);
  channel_mean_wmma<HW5, 256><<<dim3((HW5 + 15) / 16, N), wave, 0, stream>>>(x5, m5);

  nms_kernel<56, 56, 5><<<N, 128, 0, stream>>>(m3, roi3, 8.0f, 64.0f, 0.05f);
  nms_kernel<28, 28, 3><<<N, 128, 0, stream>>>(m4, roi4, 16.0f, 128.0f, 0.05f);
  nms_kernel<14, 14, 1><<<N, 128, 0, stream>>>(m5, roi5, 32.0f, 256.0f, 0.05f);

  union_kernel<<<1, 32, 0, stream>>>(roi3, roi4, roi5, ub, N);

  crop_kernel<56, 56><<<dim3(C, N), 256, 0, stream>>>(x2, ub, crop, C);
}